// GCN_layer_9345848836219
// MI455X (gfx1250) — compile-verified
//
#include <hip/hip_runtime.h>
#include <hip/hip_bf16.h>

#define D_IN  128
#define D_OUT 64

typedef float v2f __attribute__((ext_vector_type(2)));
typedef float v8f __attribute__((ext_vector_type(8)));

// ---------------------------------------------------------------------------
// Kernel 1: h = x @ W (no relu) with V_WMMA_F32_16X16X4_F32.
// Identity: relu((A x) W) == relu(A (x W)) since segment_sum is linear, so we
// transform FIRST (128 -> 64 features), halving all scatter traffic.
//
// One wave per 16-row M-tile, all 64 output columns (4 accumulators).
// Weight staged in LDS in K-PAIR-INTERLEAVED form:
//    sw2[(k/2)*64 + n] = { W[k][n], W[k+1][n] }   (32 KB)
// so each lane's B operand {B[ka][n], B[ka+1][n]} is ONE aligned ds_load_b64
// straight into an even VGPR pair -> no repacking movs before v_wmma.
//
// A 16x4 f32 layout:  lane = {m = lane&15, khalf = lane>>4};
//   VGPR0/1 = A[m][k0+2*khalf .. +1]  (contiguous float2 -> global_load_b64)
// B 4x16 f32 layout:  lane = {n = lane&15, khalf = lane>>4};
//   VGPR0/1 = B[k0+2*khalf][n], B[k0+2*khalf+1][n]
// C/D: VGPR r -> row m0 + 8*(lane>>4) + r, col = lane&15.
// ---------------------------------------------------------------------------
__global__ void gcn_xform_kernel(const float* __restrict__ x,
                                 const float* __restrict__ wgt,
                                 float*       __restrict__ h,
                                 int M) {
    __shared__ v2f sw2[(D_IN / 2) * D_OUT];     // 64 pairs x 64 cols = 32 KB

    // cooperative stage: pair p = {kp = p>>6, n = p&63}
    for (int p = threadIdx.x; p < (D_IN / 2) * D_OUT; p += blockDim.x) {
        int kp = p >> 6;
        int n  = p & 63;
        v2f v;
        v.x = wgt[(2 * kp)     * D_OUT + n];
        v.y = wgt[(2 * kp + 1) * D_OUT + n];
        sw2[p] = v;                              // ds_store_b64
    }
    __syncthreads();                             // all waves reach the barrier

    int lane  = threadIdx.x & 31;
    int mtile = (int)(blockIdx.x * (blockDim.x >> 5) + (threadIdx.x >> 5));
    int mtile_count = M >> 4;                    // 100000/16 = 6250, exact
    if (mtile < mtile_count) {                   // wave-uniform: EXEC all-1s
        int m0    = mtile << 4;
        int mrow  = lane & 15;
        int khalf = lane >> 4;                   // 0 or 1
        int ncol  = lane & 15;

        v8f c0 = {}, c1 = {}, c2 = {}, c3 = {};
        const float* arow = x + (long long)(m0 + mrow) * D_IN;

        for (int k0 = 0; k0 < D_IN; k0 += 4) {
            int ka = k0 + 2 * khalf;             // always even
            int kp = ka >> 1;

            v2f a = *(const v2f*)(arow + ka);    // global_load_b64

            const v2f* bbase = sw2 + kp * D_OUT + ncol;
            c0 = __builtin_amdgcn_wmma_f32_16x16x4_f32(false, a, false, bbase[0],
                                                       (short)0, c0, false, false);
            c1 = __builtin_amdgcn_wmma_f32_16x16x4_f32(false, a, false, bbase[16],
                                                       (short)0, c1, false, false);
            c2 = __builtin_amdgcn_wmma_f32_16x16x4_f32(false, a, false, bbase[32],
                                                       (short)0, c2, false, false);
            c3 = __builtin_amdgcn_wmma_f32_16x16x4_f32(false, a, false, bbase[48],
                                                       (short)0, c3, false, false);
        }

        int mhi = (lane >> 4) * 8;
        #pragma unroll
        for (int r = 0; r < 8; ++r) {
            float* hrow = h + (long long)(m0 + mhi + r) * D_OUT + ncol;
            hrow[0]  = c0[r];
            hrow[16] = c1[r];
            hrow[32] = c2[r];
            hrow[48] = c3[r];
        }
    }
}

// ---------------------------------------------------------------------------
// Kernel 2: zero y (output accumulator) — float4 stores
// ---------------------------------------------------------------------------
__global__ void gcn_zero_kernel(float4* __restrict__ p, long long n4) {
    long long i = (long long)blockIdx.x * blockDim.x + threadIdx.x;
    long long stride = (long long)gridDim.x * blockDim.x;
    float4 z = make_float4(0.f, 0.f, 0.f, 0.f);
    for (; i < n4; i += stride) p[i] = z;
}

// ---------------------------------------------------------------------------
// Kernel 3: edge scatter on TRANSFORMED features (64 wide).
// TWO edges per wave32: lanes 0-15 handle edge e0, lanes 16-31 edge e0+1.
// Each lane: float4 (b128) gather of h[edge_dst] (256B contiguous per
// half-wave), scale by edge_weight, 4x hardware global_atomic_add_f32 into
// y[edge_src].  h (25.6MB) and y (25.6MB) are L2-resident -> atomics resolve
// in L2.  102.4M atomics total (half of scatter-then-GEMM ordering).
// ---------------------------------------------------------------------------
__global__ void gcn_scatter_kernel(const float* __restrict__ h,
                                   const float* __restrict__ ew,
                                   const int*   __restrict__ esrc,
                                   const int*   __restrict__ edst,
                                   float*       __restrict__ y,
                                   int E) {
    int wave   = (int)((blockIdx.x * blockDim.x + threadIdx.x) >> 5);
    int lane   = threadIdx.x & 31;
    int nwaves = (int)((gridDim.x * blockDim.x) >> 5);

    int half = lane >> 4;                 // which edge of the pair
    int l    = lane & 15;                 // feature lane within half-wave

    for (int e0 = wave * 2; e0 < E; e0 += nwaves * 2) {
        // prefetch streamed index/weight arrays one sweep ahead
        if (e0 + nwaves * 2 < E) {
            __builtin_prefetch(&edst[e0 + nwaves * 2], 0, 0);
            __builtin_prefetch(&esrc[e0 + nwaves * 2], 0, 0);
            __builtin_prefetch(&ew[e0 + nwaves * 2], 0, 0);
        }

        int e = e0 + half;
        if (e < E) {                      // tail guard (E even -> always true)
            int   d = edst[e];            // gather row
            int   s = esrc[e];            // scatter row
            float w = ew[e];

            const float4* hp = (const float4*)(h + (long long)d * D_OUT) + l;
            float4 v = *hp;               // global_load_b128

            float* yp = y + (long long)s * D_OUT + l * 4;
            unsafeAtomicAdd(yp + 0, v.x * w);   // global_atomic_add_f32
            unsafeAtomicAdd(yp + 1, v.y * w);
            unsafeAtomicAdd(yp + 2, v.z * w);
            unsafeAtomicAdd(yp + 3, v.w * w);
        }
    }
}

// ---------------------------------------------------------------------------
// Kernel 4: in-place ReLU over y
// ---------------------------------------------------------------------------
__global__ void gcn_relu_kernel(float4* __restrict__ p, long long n4) {
    long long i = (long long)blockIdx.x * blockDim.x + threadIdx.x;
    long long stride = (long long)gridDim.x * blockDim.x;
    for (; i < n4; i += stride) {
        float4 v = p[i];
        v.x = fmaxf(v.x, 0.f);
        v.y = fmaxf(v.y, 0.f);
        v.z = fmaxf(v.z, 0.f);
        v.w = fmaxf(v.w, 0.f);
        p[i] = v;
    }
}

// ---------------------------------------------------------------------------
// Launch: inputs in setup_inputs() order:
//   d_in[0] = x          [N, 128]  f32
//   d_in[1] = edge_weight[E]       f32
//   d_in[2] = weight     [128, 64] f32
//   d_in[3] = edge_src   [E]       int
//   d_in[4] = edge_dst   [E]       int
// d_out = y [N, 64] f32.  d_ws holds h = x@W [N, 64] f32 (25.6 MB).
// ---------------------------------------------------------------------------
extern "C" void kernel_launch(void* const* d_in, const int* in_sizes, int n_in,
                              void* d_out, int out_size, void* d_ws, size_t ws_size,
                              hipStream_t stream) {
    const float* x    = (const float*)d_in[0];
    const float* ew   = (const float*)d_in[1];
    const float* wgt  = (const float*)d_in[2];
    const int*   esrc = (const int*)d_in[3];
    const int*   edst = (const int*)d_in[4];
    float*       y    = (float*)d_out;
    float*       h    = (float*)d_ws;

    const int N = in_sizes[0] / D_IN;    // 100000
    const int E = in_sizes[1];           // 1600000

    // 1) h = x @ W  (WMMA f32): 6250 M-tiles, 8 waves per 256-thread block
    int mtiles = N >> 4;
    int blocks = (mtiles + 7) / 8;
    gcn_xform_kernel<<<blocks, 256, 0, stream>>>(x, wgt, h, N);

    // 2) zero y accumulator (deterministic across graph replays)
    long long n4 = (long long)N * D_OUT / 4;
    gcn_zero_kernel<<<2048, 256, 0, stream>>>((float4*)y, n4);

    // 3) edge scatter on 64-wide features (2 edges per wave32 per step)
    gcn_scatter_kernel<<<4096, 256, 0, stream>>>(h, ew, esrc, edst, y, E);

    // 4) ReLU in place
    gcn_relu_kernel<<<2048, 256, 0, stream>>>((float4*)y, n4);
}